// Generator_31533649887781
// MI455X (gfx1250) — compile-verified
//
#include <hip/hip_runtime.h>
#include <hip/hip_bf16.h>

typedef __attribute__((ext_vector_type(2))) float v2f;
typedef __attribute__((ext_vector_type(8))) float v8f;
typedef __attribute__((ext_vector_type(4))) unsigned int u32x4;
typedef __attribute__((ext_vector_type(8))) int i32x8;
typedef __attribute__((ext_vector_type(4))) int i32x4;

#define B_ROWS   128
#define D_DIM    128
#define F_DIM    13
#define N_NODES  150
#define P_PAIRS  11175   // 150*149/2

// workspace layout (floats)
#define WS_NF    0                    // 128*13 = 1664
#define WS_POS   1664                 // 128*3  = 384
#define WS_BOND  2048                 // 128*4  = 512 (16B aligned)

// output layout (floats)
#define NODES_ELEMS  (B_ROWS * N_NODES * F_DIM)   // 249600
#define POS_ELEMS    (B_ROWS * N_NODES * 3)       // 57600
#define BOND_ELEMS   (B_ROWS * P_PAIRS * 4)       // 5721600
#define NODES4       (NODES_ELEMS / 4)            // 62400
#define POS4         (POS_ELEMS / 4)              // 14400
#define BOND4        (BOND_ELEMS / 4)             // 1430400
#define TOT4         (NODES4 + POS4 + BOND4)      // 1507200

static __device__ __forceinline__ v8f wmma_f32_k4(v2f a, v2f b, v8f c) {
  return __builtin_amdgcn_wmma_f32_16x16x4_f32(false, a, false, b, (short)0, c,
                                               false, false);
}

// ---------------------------------------------------------------------------
// Kernel A: all the math, one workgroup of 256 threads (8 wave32).
// All inner WMMA loops are branch-free: padded/transposed B tiles are staged
// in LDS once, so operand fetches are plain ds_load_b64 / global_load_b64.
// ---------------------------------------------------------------------------
__global__ __launch_bounds__(256) void gen_compute_kernel(
    const float* __restrict__ A,    // [128,128] encoded_protein
    const float* __restrict__ Wn,   // [128,13]
    const float* __restrict__ bn,   // [13]
    const float* __restrict__ Wp1,  // [141,128]
    const float* __restrict__ bp1,  // [128]
    const float* __restrict__ Wp2,  // [128,3]
    const float* __restrict__ bp2,  // [3]
    const float* __restrict__ We1,  // [26,128]
    const float* __restrict__ be1,  // [128]
    const float* __restrict__ We2,  // [128,4]
    const float* __restrict__ be2,  // [4]
    float* __restrict__ ws) {
  __shared__ float nf[B_ROWS * 16];        // node_feat padded to 16 cols (8 KB)
  __shared__ float posacc[B_ROWS * 4];     // position accumulator (2 KB)
  __shared__ float logacc[B_ROWS * 4];     // edge-logit accumulator (2 KB)
  __shared__ float wn_raw[B_ROWS * F_DIM]; // raw W_node via TDM (6.5 KB)
  __shared__ float wnT[16 * 128];          // W_node^T, zero-padded (8 KB)
  __shared__ float wtT[128 * 16];          // W_pos1[128:141]^T padded (8 KB)
  __shared__ float wsT[128 * 16];          // (We1[:13]+We1[13:26])^T padded (8 KB)

  const int tid  = threadIdx.x;
  const int wave = tid >> 5;        // 0..7
  const int lane = tid & 31;
  const int nl   = lane & 15;       // n for B/C/D layouts, m for A layout
  const int half = lane >> 4;       // khalf for A/B, mhalf for C/D

  // hint: pull the weight matrices toward the caches early
  __builtin_prefetch(Wp1 + (tid << 6), 0, 1);   // 141*128 f32
  __builtin_prefetch(We1 + (tid << 4), 0, 1);   // 26*128 f32

  // ---- phase 0a: TDM async-load raw W_node [128,13] f32 into LDS ---------
#if __has_builtin(__builtin_amdgcn_tensor_load_to_lds)
  if (wave == 0) {
    const unsigned long long ga = (unsigned long long)(uintptr_t)Wn;
    const unsigned int lds_addr = (unsigned int)(uintptr_t)wn_raw;
    u32x4 g0;
    g0[0] = 1u;                                   // count=1, user descriptor
    g0[1] = lds_addr;                             // LDS byte address
    g0[2] = (unsigned int)ga;                     // global_addr[31:0]
    g0[3] = (unsigned int)((ga >> 32) & 0x01ffffffu) | (2u << 30); // type=2
    i32x8 g1;
    g1[0] = (int)(2u << 16);    // data_size=2 (4 bytes), no multicast/pad
    g1[1] = (int)(13u << 16);   // tensor_dim0 = 13 (bits 63:48)
    g1[2] = (int)(128u << 16);  // tensor_dim1 = 128 (bits 95:80)
    g1[3] = (int)(13u << 16);   // tile_dim0 = 13 (bits 127:112)
    g1[4] = 128;                // tile_dim1 = 128 (bits 143:128)
    g1[5] = 13;                 // tensor_dim0_stride = 13 (bits 191:160)
    g1[6] = 0;
    g1[7] = 0;
    const i32x4 gz4 = {0, 0, 0, 0};
    const i32x8 gz8 = {0, 0, 0, 0, 0, 0, 0, 0};
    // 6-arg form (amdgpu-toolchain / clang-23 therock-10.0 headers)
    __builtin_amdgcn_tensor_load_to_lds(g0, g1, gz4, gz4, gz8, 0);
#if __has_builtin(__builtin_amdgcn_s_wait_tensorcnt)
    __builtin_amdgcn_s_wait_tensorcnt(0);
#endif
  }
#else
  for (int i = tid; i < B_ROWS * F_DIM; i += 256) wn_raw[i] = Wn[i];
#endif

  // ---- phase 0b: init accumulators, stage padded/transposed B tiles ------
  for (int i = tid; i < B_ROWS * 4; i += 256) {
    const int c = i & 3;
    posacc[i] = (c < 3) ? bp2[c] : 0.0f;
    logacc[i] = be2[c];
  }
  // wtT[n][k] = (k<13) ? Wp1[128+k][n] : 0      (128 x 16)
  // wsT[n][k] = (k<13) ? We1[k][n]+We1[k+13][n] : 0
  for (int i = tid; i < 128 * 16; i += 256) {
    const int n = i >> 4;
    const int k = i & 15;
    wtT[i] = (k < F_DIM) ? Wp1[(D_DIM + k) * 128 + n] : 0.0f;
    wsT[i] = (k < F_DIM)
                 ? (We1[k * 128 + n] + We1[(k + F_DIM) * 128 + n])
                 : 0.0f;
  }
  __syncthreads();   // wn_raw (TDM) + wtT/wsT visible

  // wnT[n][k] = (n<13) ? Wn[k][n] : 0           (16 x 128)
  for (int i = tid; i < 16 * 128; i += 256) {
    const int n = i >> 7;
    const int k = i & 127;
    wnT[i] = (n < F_DIM) ? wn_raw[k * F_DIM + n] : 0.0f;
  }
  __syncthreads();

  // ---- phase 1: node_feat = A @ W_node + b_node  (wave w -> M tile w) ----
  {
    const int m0 = wave * 16;
    v8f acc = {};
    for (int kk = 0; kk < 32; ++kk) {
      const int kb = kk * 4 + half * 2;
      const v2f a = *(const v2f*)(A + (m0 + nl) * D_DIM + kb);   // b64
      const v2f b = *(const v2f*)(wnT + nl * 128 + kb);          // ds b64
      acc = wmma_f32_k4(a, b, acc);
    }
    const float bias = (nl < F_DIM) ? bn[nl] : 0.0f;
#pragma unroll
    for (int r = 0; r < 8; ++r) {
      const int m = m0 + r + 8 * half;
      const float v = acc[r] + bias;           // padded cols stay exactly 0
      nf[m * 16 + nl] = v;
      if (nl < F_DIM) ws[WS_NF + m * F_DIM + nl] = v;
    }
  }
  __syncthreads();

  // ---- phase 2: pos_hidden tiles + fused @W_pos2 reduction ---------------
  // pos_in @ W_pos1 = A @ W_pos1[0:128] + nf @ W_pos1[128:141]
  for (int t = wave; t < 64; t += 8) {
    const int mt = (t >> 3) * 16;
    const int nt = (t & 7) * 16;
    const int ncol = nt + nl;
    v8f acc = {};
    for (int kk = 0; kk < 32; ++kk) {          // dense K over A
      const int kb = kk * 4 + half * 2;
      const v2f a = *(const v2f*)(A + (mt + nl) * D_DIM + kb);
      v2f b;
      b.x = Wp1[kb * 128 + ncol];
      b.y = Wp1[(kb + 1) * 128 + ncol];
      acc = wmma_f32_k4(a, b, acc);
    }
    for (int kk = 0; kk < 4; ++kk) {           // K over node_feat (13 pad 16)
      const int kb = kk * 4 + half * 2;
      const v2f a = *(const v2f*)(nf + (mt + nl) * 16 + kb);
      const v2f b = *(const v2f*)(wtT + ncol * 16 + kb);
      acc = wmma_f32_k4(a, b, acc);
    }
    const float bias = bp1[ncol];
    const float w0 = Wp2[ncol * 3 + 0];
    const float w1 = Wp2[ncol * 3 + 1];
    const float w2 = Wp2[ncol * 3 + 2];
#pragma unroll
    for (int r = 0; r < 8; ++r) {
      const int m = mt + r + 8 * half;
      const float v = fmaxf(acc[r] + bias, 0.0f);
      atomicAdd(&posacc[m * 4 + 0], v * w0);   // ds_add_f32
      atomicAdd(&posacc[m * 4 + 1], v * w1);
      atomicAdd(&posacc[m * 4 + 2], v * w2);
    }
  }

  // ---- phase 3: h = relu(nf @ Wsum + b) fused with @W_edge2 --------------
  for (int t = wave; t < 64; t += 8) {
    const int mt = (t >> 3) * 16;
    const int nt = (t & 7) * 16;
    const int ncol = nt + nl;
    v8f acc = {};
    for (int kk = 0; kk < 4; ++kk) {
      const int kb = kk * 4 + half * 2;
      const v2f a = *(const v2f*)(nf + (mt + nl) * 16 + kb);
      const v2f b = *(const v2f*)(wsT + ncol * 16 + kb);
      acc = wmma_f32_k4(a, b, acc);
    }
    const float bias = be1[ncol];
    const float e0 = We2[ncol * 4 + 0];
    const float e1 = We2[ncol * 4 + 1];
    const float e2 = We2[ncol * 4 + 2];
    const float e3 = We2[ncol * 4 + 3];
#pragma unroll
    for (int r = 0; r < 8; ++r) {
      const int m = mt + r + 8 * half;
      const float v = fmaxf(acc[r] + bias, 0.0f);
      atomicAdd(&logacc[m * 4 + 0], v * e0);
      atomicAdd(&logacc[m * 4 + 1], v * e1);
      atomicAdd(&logacc[m * 4 + 2], v * e2);
      atomicAdd(&logacc[m * 4 + 3], v * e3);
    }
  }
  __syncthreads();

  // ---- phase 4: emit position + softmax(bond logits) ---------------------
  if (tid < B_ROWS) {
    const int m = tid;
    ws[WS_POS + m * 3 + 0] = posacc[m * 4 + 0];
    ws[WS_POS + m * 3 + 1] = posacc[m * 4 + 1];
    ws[WS_POS + m * 3 + 2] = posacc[m * 4 + 2];

    const float l0 = logacc[m * 4 + 0];
    const float l1 = logacc[m * 4 + 1];
    const float l2 = logacc[m * 4 + 2];
    const float l3 = logacc[m * 4 + 3];
    const float mx = fmaxf(fmaxf(l0, l1), fmaxf(l2, l3));
    const float x0 = __expf(l0 - mx);
    const float x1 = __expf(l1 - mx);
    const float x2 = __expf(l2 - mx);
    const float x3 = __expf(l3 - mx);
    const float inv = 1.0f / (x0 + x1 + x2 + x3);
    ws[WS_BOND + m * 4 + 0] = x0 * inv;
    ws[WS_BOND + m * 4 + 1] = x1 * inv;
    ws[WS_BOND + m * 4 + 2] = x2 * inv;
    ws[WS_BOND + m * 4 + 3] = x3 * inv;
  }
}

// ---------------------------------------------------------------------------
// Kernel B: HBM-saturating broadcast of the 24.1 MB output via b128 stores.
// ---------------------------------------------------------------------------
__global__ __launch_bounds__(256) void gen_broadcast_kernel(
    const float* __restrict__ ws, float4* __restrict__ out4) {
  int t = blockIdx.x * blockDim.x + threadIdx.x;
  const int stride = gridDim.x * blockDim.x;
  const float* nf = ws + WS_NF;
  const float* pos = ws + WS_POS;
  const float4* bond4 = (const float4*)(ws + WS_BOND);  // 128 x float4

  __builtin_prefetch(ws + ((t & 127) << 4), 0, 1);  // keep ws hot in-cache

  for (; t < TOT4; t += stride) {
    float4 o;
    if (t < NODES4) {
      // nodes: per-b block of 150*13 = 1950 floats, value = nf[b*13 + f]
      const int i = t * 4;
      float c[4];
#pragma unroll
      for (int j = 0; j < 4; ++j) {
        const int k = i + j;
        const int b = k / 1950;
        const int f = (k % 1950) % F_DIM;
        c[j] = nf[b * F_DIM + f];
      }
      o = make_float4(c[0], c[1], c[2], c[3]);
    } else if (t < NODES4 + POS4) {
      // positions: per-b block of 150*3 = 450 floats, value = pos[b*3 + f]
      const int i = (t - NODES4) * 4;
      float c[4];
#pragma unroll
      for (int j = 0; j < 4; ++j) {
        const int k = i + j;
        const int b = k / 450;
        const int f = (k % 450) % 3;
        c[j] = pos[b * 3 + f];
      }
      o = make_float4(c[0], c[1], c[2], c[3]);
    } else {
      // bond: per-b block of 11175*4 floats; 44700 % 4 == 0, so one float4
      // in == one float4 out. This region is 95% of all traffic.
      const int k4 = t - (NODES4 + POS4);
      const int b = k4 / P_PAIRS;
      o = bond4[b];   // L2-resident b128 load
    }
    out4[t] = o;      // global_store_b128
  }
}

// ---------------------------------------------------------------------------
extern "C" void kernel_launch(void* const* d_in, const int* in_sizes, int n_in,
                              void* d_out, int out_size, void* d_ws,
                              size_t ws_size, hipStream_t stream) {
  const float* A   = (const float*)d_in[0];
  const float* Wn  = (const float*)d_in[1];
  const float* bn  = (const float*)d_in[2];
  const float* Wp1 = (const float*)d_in[3];
  const float* bp1 = (const float*)d_in[4];
  const float* Wp2 = (const float*)d_in[5];
  const float* bp2 = (const float*)d_in[6];
  const float* We1 = (const float*)d_in[7];
  const float* be1 = (const float*)d_in[8];
  const float* We2 = (const float*)d_in[9];
  const float* be2 = (const float*)d_in[10];
  float* ws = (float*)d_ws;
  float* out = (float*)d_out;

  gen_compute_kernel<<<1, 256, 0, stream>>>(A, Wn, bn, Wp1, bp1, Wp2, bp2, We1,
                                            be1, We2, be2, ws);
  const int blocks = (TOT4 + 255) / 256;  // 5888
  gen_broadcast_kernel<<<blocks, 256, 0, stream>>>(ws, (float4*)out);
}